// ProposalLayer_84387517431931
// MI455X (gfx1250) — compile-verified
//
#include <hip/hip_runtime.h>
#include <hip/hip_bf16.h>

// ---------------------------------------------------------------------------
// ProposalLayer for MI455X (gfx1250, wave32).
// No matmul exists in this op -> WMMA not applicable. CDNA5-specific paths
// used instead: async global->LDS copies (ASYNCcnt / s_wait_asynccnt),
// 320KB-WGP LDS-resident NMS state (120KB/block), wave32 shuffle reductions.
// ---------------------------------------------------------------------------

#define B_      32
#define H_      100
#define W_      100
#define A_      9
#define K_      (H_ * W_ * A_)      // 90000 anchors per batch
#define PRE_    6000
#define POST_   300
#define NBINS_  4096
#define BCAP_   4096
#define NEG_    (-1000000000.0f)
#define DEAD_   (-3.0e30f)
#define NMS_T_  0.7f
#define NMSBD_  512                 // NMS block dim (16 waves of 32)

// ---------------------------------------------------------------------------
// Box decode, identical math to the reference (fp32, expf, clip, min-size).
// Returns validity (min-size mask). i = (h*W + w)*A + a.
// ---------------------------------------------------------------------------
__device__ __forceinline__ bool decode_box(
    const float* __restrict__ cls, const float* __restrict__ bbox,
    const float* __restrict__ anc, const float* __restrict__ info,
    int b, int i,
    float& s, float& x1, float& y1, float& x2, float& y2)
{
    const int HW = H_ * W_;
    int a = i % A_;
    int p = i / A_;
    int w = p % W_;
    int h = p / W_;

    s = cls[(((size_t)b * 2 * A_ + A_ + a) * H_ + h) * W_ + w];

    size_t base = (((size_t)b * 4 * A_) * H_ + h) * W_ + w;
    float dx = bbox[base + (size_t)(4 * a + 0) * HW];
    float dy = bbox[base + (size_t)(4 * a + 1) * HW];
    float dw = bbox[base + (size_t)(4 * a + 2) * HW];
    float dh = bbox[base + (size_t)(4 * a + 3) * HW];

    float ax1 = anc[a * 4 + 0] + (float)w * 16.0f;
    float ay1 = anc[a * 4 + 1] + (float)h * 16.0f;
    float ax2 = anc[a * 4 + 2] + (float)w * 16.0f;
    float ay2 = anc[a * 4 + 3] + (float)h * 16.0f;

    float wa  = ax2 - ax1 + 1.0f;
    float ha  = ay2 - ay1 + 1.0f;
    float cxa = ax1 + 0.5f * (wa - 1.0f);
    float cya = ay1 + 0.5f * (ha - 1.0f);

    float cx = dx * wa + cxa;
    float cy = dy * ha + cya;
    float pw = expf(dw) * wa;
    float ph = expf(dh) * ha;

    x1 = cx - 0.5f * (pw - 1.0f);
    y1 = cy - 0.5f * (ph - 1.0f);
    x2 = cx + 0.5f * (pw - 1.0f);
    y2 = cy + 0.5f * (ph - 1.0f);

    float imh = info[b * 3 + 0];
    float imw = info[b * 3 + 1];
    float sc  = info[b * 3 + 2];

    x1 = fminf(fmaxf(x1, 0.0f), imw - 1.0f);
    y1 = fminf(fmaxf(y1, 0.0f), imh - 1.0f);
    x2 = fminf(fmaxf(x2, 0.0f), imw - 1.0f);
    y2 = fminf(fmaxf(y2, 0.0f), imh - 1.0f);

    float msz = 16.0f * sc;
    return (x2 - x1 + 1.0f >= msz) && (y2 - y1 + 1.0f >= msz);
}

__device__ __forceinline__ int score_bin(float s)
{
    int bin = (int)(s * (float)NBINS_);
    bin = bin < 0 ? 0 : (bin > NBINS_ - 1 ? NBINS_ - 1 : bin);
    return bin;
}

// ---------------------------------------------------------------------------
// Kernel Z: zero histograms + counters (scratch is poisoned by harness).
// ---------------------------------------------------------------------------
__global__ void k_zero(int* __restrict__ hist, int* __restrict__ ccnt,
                       int* __restrict__ bcnt)
{
    int t = blockIdx.x * blockDim.x + threadIdx.x;
    if (t < B_ * NBINS_) hist[t] = 0;
    if (t < B_) { ccnt[t] = 0; bcnt[t] = 0; }
}

// ---------------------------------------------------------------------------
// Kernel A: decode + per-batch histogram of valid scores (uniform [0,1)).
// ---------------------------------------------------------------------------
__global__ void k_hist(const float* __restrict__ cls, const float* __restrict__ bbox,
                       const float* __restrict__ anc, const float* __restrict__ info,
                       int* __restrict__ hist)
{
    int t = blockIdx.x * blockDim.x + threadIdx.x;
    if (t >= B_ * K_) return;
    int b = t / K_;
    int i = t - b * K_;
    float s, x1, y1, x2, y2;
    if (!decode_box(cls, bbox, anc, info, b, i, s, x1, y1, x2, y2)) return;
    atomicAdd(&hist[b * NBINS_ + score_bin(s)], 1);
}

// ---------------------------------------------------------------------------
// Kernel B: per batch, scan histogram from top bin down to find the exact
// top-PRE_ threshold bin and how many elements of that bin are needed.
// ---------------------------------------------------------------------------
__global__ void k_scan(const int* __restrict__ hist, int* __restrict__ thrbin,
                       int* __restrict__ needv)
{
    int b = threadIdx.x;
    if (b >= B_) return;
    int cum = 0, t = -1, need = 0;
    for (int bin = NBINS_ - 1; bin >= 0; --bin) {
        int c = hist[b * NBINS_ + bin];
        if (cum + c >= PRE_) { t = bin; need = PRE_ - cum; break; }
        cum += c;
    }
    thrbin[b] = t;   // -1 => fewer than PRE_ valid boxes: take everything
    needv[b]  = need;
}

// ---------------------------------------------------------------------------
// Kernel C: second decode pass. Strictly-above-threshold -> candidate list,
// boundary bin -> side list (with original index for tie-breaking).
// ---------------------------------------------------------------------------
__global__ void k_collect(const float* __restrict__ cls, const float* __restrict__ bbox,
                          const float* __restrict__ anc, const float* __restrict__ info,
                          const int* __restrict__ thrbin,
                          int* __restrict__ ccnt, int* __restrict__ bcnt,
                          float* __restrict__ c_score, float4* __restrict__ c_box,
                          float* __restrict__ b_score, float4* __restrict__ b_box,
                          int* __restrict__ b_idx)
{
    int t = blockIdx.x * blockDim.x + threadIdx.x;
    if (t >= B_ * K_) return;
    int b = t / K_;
    int i = t - b * K_;
    float s, x1, y1, x2, y2;
    if (!decode_box(cls, bbox, anc, info, b, i, s, x1, y1, x2, y2)) return;
    int bin = score_bin(s);
    int thr = thrbin[b];
    if (bin > thr) {
        int pos = atomicAdd(&ccnt[b], 1);
        if (pos < PRE_) {
            c_score[(size_t)b * PRE_ + pos] = s;
            c_box[(size_t)b * PRE_ + pos]   = make_float4(x1, y1, x2, y2);
        }
    } else if (bin == thr) {
        int pos = atomicAdd(&bcnt[b], 1);
        if (pos < BCAP_) {
            b_score[(size_t)b * BCAP_ + pos] = s;
            b_box[(size_t)b * BCAP_ + pos]   = make_float4(x1, y1, x2, y2);
            b_idx[(size_t)b * BCAP_ + pos]   = i;
        }
    }
}

// ---------------------------------------------------------------------------
// Kernel D: exact trim of the boundary bin. Rank by (score desc, index asc);
// append the top `need` to the candidate list. C is typically ~22.
// ---------------------------------------------------------------------------
__global__ void k_trim(const int* __restrict__ needv,
                       int* __restrict__ ccnt, const int* __restrict__ bcnt,
                       const float* __restrict__ b_score, const float4* __restrict__ b_box,
                       const int* __restrict__ b_idx,
                       float* __restrict__ c_score, float4* __restrict__ c_box)
{
    int b = blockIdx.x;
    int C = bcnt[b]; if (C > BCAP_) C = BCAP_;
    int need = needv[b];
    if (need <= 0 || C <= 0) return;

    const float*  bs = b_score + (size_t)b * BCAP_;
    const float4* bb = b_box   + (size_t)b * BCAP_;
    const int*    bi = b_idx   + (size_t)b * BCAP_;
    bool take_all = (C <= need);

    for (int i = threadIdx.x; i < C; i += blockDim.x) {
        bool take = true;
        if (!take_all) {
            float si = bs[i];
            int   ii = bi[i];
            int rank = 0;
            for (int j = 0; j < C; ++j) {
                float sj = bs[j];
                if (sj > si || (sj == si && bi[j] < ii)) rank++;
            }
            take = (rank < need);
        }
        if (take) {
            int pos = atomicAdd(&ccnt[b], 1);
            if (pos < PRE_) {
                c_score[(size_t)b * PRE_ + pos] = bs[i];
                c_box[(size_t)b * PRE_ + pos]   = bb[i];
            }
        }
    }
}

// ---------------------------------------------------------------------------
// Kernel E: greedy NMS, one block per batch. Candidate scores AND boxes are
// staged into LDS (120 KB of the WGP's 320 KB) with CDNA5 async global->LDS
// copies (b32 + b128, ASYNCcnt). The 300 sequential argmax/suppress rounds
// then run entirely out of LDS: zero VMEM on the critical path.
// ---------------------------------------------------------------------------
__global__ __launch_bounds__(NMSBD_) void k_nms(
    const float* __restrict__ c_score, const float4* __restrict__ c_box,
    const int* __restrict__ ccnt, float* __restrict__ out)
{
    __shared__ float  lscore[PRE_];         // 24 KB
    __shared__ float4 lbox[PRE_];           // 96 KB
    __shared__ float  rv[NMSBD_ / 32];
    __shared__ int    ri[NMSBD_ / 32];
    __shared__ int    sj;
    __shared__ float  sb[4];

    int b   = blockIdx.x;
    int tid = threadIdx.x;
    int n   = ccnt[b]; if (n > PRE_) n = PRE_;

    const float*  gs = c_score + (size_t)b * PRE_;
    const float4* gb = c_box   + (size_t)b * PRE_;

    // --- async global->LDS fill of candidate scores + boxes ---------------
    // Generic LDS pointer low 32 bits == LDS byte address (ISA 10.2).
    for (int i = tid; i < n; i += NMSBD_) {
        unsigned lds_s         = (unsigned)(unsigned long long)(const void*)&lscore[i];
        unsigned long long gas = (unsigned long long)(const void*)(gs + i);
        asm volatile("global_load_async_to_lds_b32 %0, %1, off"
                     :: "v"(lds_s), "v"(gas) : "memory");
        unsigned lds_b         = (unsigned)(unsigned long long)(const void*)&lbox[i];
        unsigned long long gab = (unsigned long long)(const void*)(gb + i);
        asm volatile("global_load_async_to_lds_b128 %0, %1, off"
                     :: "v"(lds_b), "v"(gab) : "memory");
    }
#if __has_builtin(__builtin_amdgcn_s_wait_asynccnt)
    __builtin_amdgcn_s_wait_asynccnt(0);
#else
    asm volatile("s_wait_asynccnt 0" ::: "memory");
#endif
    __syncthreads();

    for (int sel = 0; sel < POST_; ++sel) {
        // ---- block argmax over live scores (LDS only) --------------------
        float mv = DEAD_;
        int   mi = -1;
        for (int k = tid; k < n; k += NMSBD_) {
            float v = lscore[k];
            if (v > mv) { mv = v; mi = k; }
        }
        for (int off = 16; off > 0; off >>= 1) {        // wave32 reduce
            float ov = __shfl_down(mv, off, 32);
            int   oi = __shfl_down(mi, off, 32);
            if (ov > mv) { mv = ov; mi = oi; }
        }
        if ((tid & 31) == 0) { rv[tid >> 5] = mv; ri[tid >> 5] = mi; }
        __syncthreads();

        if (tid == 0) {
            float bv = DEAD_;
            int   bi2 = -1;
            for (int w = 0; w < NMSBD_ / 32; ++w)
                if (rv[w] > bv) { bv = rv[w]; bi2 = ri[w]; }
            int j = (bv > NEG_ * 0.5f) ? bi2 : -1;      // validity check
            sj = j;
            if (j >= 0) {
                float4 bx = lbox[j];
                sb[0] = bx.x; sb[1] = bx.y; sb[2] = bx.z; sb[3] = bx.w;
                lscore[j] = DEAD_;
                float* orow = out + ((size_t)b * POST_ + sel) * 5;
                orow[0] = (float)b;
                orow[1] = bx.x; orow[2] = bx.y; orow[3] = bx.z; orow[4] = bx.w;
            }
        }
        __syncthreads();

        int j = sj;
        if (j < 0) {
            // no live candidates left: rows sel..POST-1 are [b,0,0,0,0]
            for (int q = tid; q < (POST_ - sel) * 5; q += NMSBD_) {
                int r = sel + q / 5;
                int c = q - (q / 5) * 5;
                out[((size_t)b * POST_ + r) * 5 + c] = (c == 0) ? (float)b : 0.0f;
            }
            break;
        }

        // ---- suppress by IoU against selected box (LDS only) -------------
        float jx1 = sb[0], jy1 = sb[1], jx2 = sb[2], jy2 = sb[3];
        float areaj = (jx2 - jx1 + 1.0f) * (jy2 - jy1 + 1.0f);
        for (int k = tid; k < n; k += NMSBD_) {
            if (lscore[k] <= DEAD_ * 0.5f) continue;    // already dead
            float4 bk = lbox[k];
            float xx1 = fmaxf(jx1, bk.x), yy1 = fmaxf(jy1, bk.y);
            float xx2 = fminf(jx2, bk.z), yy2 = fminf(jy2, bk.w);
            float iw = fmaxf(xx2 - xx1 + 1.0f, 0.0f);
            float ih = fmaxf(yy2 - yy1 + 1.0f, 0.0f);
            float inter = iw * ih;
            float areak = (bk.z - bk.x + 1.0f) * (bk.w - bk.y + 1.0f);
            float iou = inter / (areaj + areak - inter);
            if (iou > NMS_T_) lscore[k] = DEAD_;
        }
        __syncthreads();
    }
}

// ---------------------------------------------------------------------------
// Launch
// ---------------------------------------------------------------------------
extern "C" void kernel_launch(void* const* d_in, const int* in_sizes, int n_in,
                              void* d_out, int out_size, void* d_ws, size_t ws_size,
                              hipStream_t stream)
{
    const float* cls  = (const float*)d_in[0];   // (32, 18, 100, 100)
    const float* bbox = (const float*)d_in[1];   // (32, 36, 100, 100)
    const float* info = (const float*)d_in[2];   // (32, 3)
    const float* anc  = (const float*)d_in[3];   // (9, 4)
    float* out = (float*)d_out;                  // (32, 300, 5)

    // --- scratch partition (~7.5 MB), float4 arrays first for alignment ---
    char* p = (char*)d_ws;
    float4* c_box   = (float4*)p; p += (size_t)B_ * PRE_  * sizeof(float4);
    float4* b_box   = (float4*)p; p += (size_t)B_ * BCAP_ * sizeof(float4);
    float*  c_score = (float*)p;  p += (size_t)B_ * PRE_  * sizeof(float);
    float*  b_score = (float*)p;  p += (size_t)B_ * BCAP_ * sizeof(float);
    int*    b_idx   = (int*)p;    p += (size_t)B_ * BCAP_ * sizeof(int);
    int*    hist    = (int*)p;    p += (size_t)B_ * NBINS_ * sizeof(int);
    int*    thrbin  = (int*)p;    p += B_ * sizeof(int);
    int*    needv   = (int*)p;    p += B_ * sizeof(int);
    int*    ccnt    = (int*)p;    p += B_ * sizeof(int);
    int*    bcnt    = (int*)p;    p += B_ * sizeof(int);

    const int NTOT = B_ * K_;

    k_zero<<<(B_ * NBINS_ + 255) / 256, 256, 0, stream>>>(hist, ccnt, bcnt);
    k_hist<<<(NTOT + 255) / 256, 256, 0, stream>>>(cls, bbox, anc, info, hist);
    k_scan<<<1, 32, 0, stream>>>(hist, thrbin, needv);
    k_collect<<<(NTOT + 255) / 256, 256, 0, stream>>>(cls, bbox, anc, info, thrbin,
                                                      ccnt, bcnt, c_score, c_box,
                                                      b_score, b_box, b_idx);
    k_trim<<<B_, 256, 0, stream>>>(needv, ccnt, bcnt, b_score, b_box, b_idx,
                                   c_score, c_box);
    k_nms<<<B_, NMSBD_, 0, stream>>>(c_score, c_box, ccnt, out);
}